// LogicLayer_41223096107632
// MI455X (gfx1250) — compile-verified
//
#include <hip/hip_runtime.h>
#include <cstdint>

static constexpr int kInDim   = 4096;
static constexpr int kOutDim  = 4096;
static constexpr int kBatch   = 16384;
static constexpr int kRows    = 4;     // batch rows staged in LDS per block (4*16KB = 64KB)
static constexpr int kThreads = 256;   // 8 waves (wave32)

typedef __attribute__((ext_vector_type(2))) float v2f;
typedef __attribute__((ext_vector_type(8))) float v8f;

__constant__ float c_gate[16][4] = {
    {0,0,0,0},  {0,0,0,1},  {0,1,0,-1}, {0,1,0,0},
    {0,0,1,-1}, {0,0,1,0},  {0,1,1,-2}, {0,1,1,-1},
    {1,-1,-1,1},{1,-1,-1,2},{1,0,-1,0}, {1,0,-1,1},
    {1,-1,0,0}, {1,-1,0,1}, {1,0,0,-1}, {1,0,0,0}};

// ---------------------------------------------------------------------------
// Kernel 1: coef = softmax(W, axis=-1) @ GATE   (4096x16 @ 16x4 -> 4096x4)
// One wave (32 lanes) per 16 output rows. The 16x16 @ 16x4 (padded to 16)
// tile product is done with 4 chained V_WMMA_F32_16X16X4_F32 ops.
// ---------------------------------------------------------------------------
__global__ __launch_bounds__(32)
void coef_kernel(const float* __restrict__ w, float4* __restrict__ coef) {
    __shared__ float P[16][17];          // softmax probabilities, padded
    const int lane = threadIdx.x;        // 0..31 (wave32)
    const int row0 = blockIdx.x * 16;

    if (lane < 16) {
        const float* wr = w + (size_t)(row0 + lane) * 16;
        float v[16];
        float mx = wr[0];
        #pragma unroll
        for (int k = 0; k < 16; ++k) { v[k] = wr[k]; mx = fmaxf(mx, v[k]); }
        float s = 0.f;
        #pragma unroll
        for (int k = 0; k < 16; ++k) { v[k] = __expf(v[k] - mx); s += v[k]; }
        const float inv = 1.0f / s;
        #pragma unroll
        for (int k = 0; k < 16; ++k) P[lane][k] = v[k] * inv;
    }
    __syncthreads();

    const int m = lane & 15;             // matrix row (A) / column (B,D)
    const int h = lane >> 4;             // half-wave select

    v8f acc = {};
    #pragma unroll
    for (int k4 = 0; k4 < 4; ++k4) {
        // A fragment (16x4 f32): V0 = K0|K2, V1 = K1|K3 per half-wave
        v2f A;
        A.x = P[m][4 * k4 + 2 * h + 0];
        A.y = P[m][4 * k4 + 2 * h + 1];
        // B fragment (4x16 f32), GATE rows 4*k4.., columns padded 4 -> 16
        v2f B;
        float bx = 0.f, by = 0.f;
        if (m < 4) {
            bx = c_gate[4 * k4 + 2 * h + 0][m];
            by = c_gate[4 * k4 + 2 * h + 1][m];
        }
        B.x = bx; B.y = by;
        acc = __builtin_amdgcn_wmma_f32_16x16x4_f32(
            false, A, false, B, (short)0, acc, false, false);
    }

    // D layout: lane (n + 16*hi) holds rows (vr + 8*hi) of column n in acc[vr]
    if (m < 4) {
        float* cf = (float*)coef;
        #pragma unroll
        for (int vr = 0; vr < 8; ++vr) {
            const int orow = row0 + vr + 8 * h;
            cf[(size_t)orow * 4 + m] = acc[vr];
        }
    }
}

// ---------------------------------------------------------------------------
// Kernel 2: main gather + gate evaluation. Stage kRows rows of x in LDS via
// CDNA5 async global->LDS loads, then each thread evaluates output columns
// j = tid, tid+256, ... for all staged rows. Gathers hit LDS, stores are
// coalesced non-temporal.
// ---------------------------------------------------------------------------
__global__ __launch_bounds__(kThreads)
void gate_kernel(const float* __restrict__ x,
                 const int*   __restrict__ idx_a,
                 const int*   __restrict__ idx_b,
                 const float4* __restrict__ coef,
                 float* __restrict__ out) {
    __shared__ float tile[kRows * kInDim];   // 64 KB

    const int  t    = threadIdx.x;
    const long row0 = (long)blockIdx.x * kRows;
    const float* __restrict__ src = x + row0 * kInDim;

    // ---- async stage: kRows*kInDim floats = 4096 b128 chunks, 16 per thread
    constexpr int kChunks = (kRows * kInDim / 4) / kThreads;
    #pragma unroll
    for (int i = 0; i < kChunks; ++i) {
        const int c = t + i * kThreads;
        const unsigned lds_off =
            (unsigned)(uintptr_t)(&tile[c * 4]);            // LDS byte offset
        const unsigned long long gaddr =
            (unsigned long long)(uintptr_t)(src + c * 4);
        asm volatile("global_load_async_to_lds_b128 %0, %1, off"
                     :: "v"(lds_off), "v"(gaddr) : "memory");
    }
    asm volatile("s_wait_asynccnt 0x0" ::: "memory");
    __syncthreads();

    // ---- gather + gate
    const long obase = row0 * (long)kOutDim;
    #pragma unroll 2
    for (int i = 0; i < kOutDim / kThreads; ++i) {
        const int j  = t + i * kThreads;
        const int ia = idx_a[j];
        const int ib = idx_b[j];
        const float4 cf = coef[j];
        #pragma unroll
        for (int r = 0; r < kRows; ++r) {
            const float a = tile[r * kInDim + ia];
            const float b = tile[r * kInDim + ib];
            const float y = fmaf(cf.w, a * b,
                            fmaf(cf.z, b,
                            fmaf(cf.y, a, cf.x)));
            __builtin_nontemporal_store(y, &out[obase + (long)r * kOutDim + j]);
        }
    }
}

// ---------------------------------------------------------------------------
extern "C" void kernel_launch(void* const* d_in, const int* in_sizes, int n_in,
                              void* d_out, int out_size, void* d_ws, size_t ws_size,
                              hipStream_t stream) {
    (void)in_sizes; (void)n_in; (void)out_size; (void)ws_size;
    const float* x     = (const float*)d_in[0];
    const float* w     = (const float*)d_in[1];
    const int*   idx_a = (const int*)d_in[2];
    const int*   idx_b = (const int*)d_in[3];
    float*       out   = (float*)d_out;
    float4*      coef  = (float4*)d_ws;    // 4096 * 16 B = 64 KB scratch

    coef_kernel<<<kOutDim / 16, 32, 0, stream>>>(w, coef);
    gate_kernel<<<kBatch / kRows, kThreads, 0, stream>>>(x, idx_a, idx_b, coef, out);
}